// TokenReorderer_28252294873409
// MI455X (gfx1250) — compile-verified
//
#include <hip/hip_runtime.h>

#define NUM_EXPERTS 8
#define TOPK 2
#define NTOK 8388608
#define MTOT (NTOK * TOPK)        // 16777216 flat (token, expert) entries
#define NBLK 2048
#define NTHR 256
#define PER_THREAD 32
#define CHUNK (NTHR * PER_THREAD) // 8192 elements per block

static_assert((size_t)NBLK * CHUNK == (size_t)MTOT, "tiling must cover input exactly");

// gfx1250 async global->LDS copy path (ASYNCcnt-tracked LDS DMA), with
// graceful fallback to synchronous b128 staging if the builtins are absent.
#if defined(__AMDGCN__) && \
    __has_builtin(__builtin_amdgcn_global_load_async_to_lds_b128) && \
    __has_builtin(__builtin_amdgcn_s_wait_asynccnt)
#define HAVE_ASYNC_LDS 1
#else
#define HAVE_ASYNC_LDS 0
#endif

typedef int v4i __attribute__((ext_vector_type(4)));
typedef __attribute__((address_space(1))) v4i* gv4i_ptr;   // global
typedef __attribute__((address_space(3))) v4i* lv4i_ptr;   // LDS

// ---------------------------------------------------------------------------
// Kernel 1: per-block expert histogram (coalesced int4 loads, packed byte
// counters in registers, LDS ds_add reduction).
// ---------------------------------------------------------------------------
__global__ __launch_bounds__(NTHR) void k_hist(const int* __restrict__ experts,
                                               int* __restrict__ counts) {
    __shared__ int h[NUM_EXPERTS];
    const int t = threadIdx.x;
    const int b = blockIdx.x;
    if (t < NUM_EXPERTS) h[t] = 0;
    __syncthreads();

    const int4* e4 = (const int4*)(experts + (size_t)b * CHUNK);
    unsigned lo = 0, hi = 0;  // 8 byte-counters (experts 0-3 in lo, 4-7 in hi)
#pragma unroll
    for (int i = 0; i < PER_THREAD / 4; ++i) {
        int4 v = e4[i * NTHR + t];
        int ev[4] = {v.x, v.y, v.z, v.w};
#pragma unroll
        for (int k = 0; k < 4; ++k) {
            unsigned e = (unsigned)ev[k];
            unsigned inc = 1u << ((e & 3u) * 8u);
            if (e & 4u) hi += inc; else lo += inc;
        }
    }
#pragma unroll
    for (int e = 0; e < 4; ++e) {
        atomicAdd(&h[e],     (int)((lo >> (e * 8)) & 0xffu));
        atomicAdd(&h[e + 4], (int)((hi >> (e * 8)) & 0xffu));
    }
    __syncthreads();
    if (t < NUM_EXPERTS) counts[b * NUM_EXPERTS + t] = h[t];
}

// ---------------------------------------------------------------------------
// Kernel 2: exclusive scan of per-block counts, per expert. One wave32 per
// expert, shuffle-based scan over NBLK entries. Also emits expert base
// offsets and num_tokens_per_expert (as float) to d_out tail.
// ---------------------------------------------------------------------------
__global__ __launch_bounds__(256) void k_scan(const int* __restrict__ counts,
                                              int* __restrict__ offsets,
                                              int* __restrict__ bases,
                                              float* __restrict__ out_counts) {
    __shared__ int tot[NUM_EXPERTS];
    __shared__ int baseS[NUM_EXPERTS];
    const int t = threadIdx.x;
    const int w = t >> 5;    // expert handled by this wave
    const int lane = t & 31;

    int carry = 0;
    for (int c = 0; c < NBLK / 32; ++c) {
        const int b = c * 32 + lane;
        const int v = counts[b * NUM_EXPERTS + w];
        int incl = v;
#pragma unroll
        for (int d = 1; d < 32; d <<= 1) {
            int n = __shfl_up(incl, d, 32);
            if (lane >= d) incl += n;
        }
        offsets[b * NUM_EXPERTS + w] = carry + incl - v;   // exclusive within expert
        carry += __shfl(incl, 31, 32);                     // uniform across lanes
    }
    if (lane == 0) tot[w] = carry;
    __syncthreads();
    if (t == 0) {
        int acc = 0;
        for (int e = 0; e < NUM_EXPERTS; ++e) { baseS[e] = acc; acc += tot[e]; }
    }
    __syncthreads();
    if (t < NUM_EXPERTS) {
        bases[t] = baseS[t];
        out_counts[t] = (float)tot[t];   // counts < 2^24 -> exact in f32
    }
}

// ---------------------------------------------------------------------------
// Kernel 3: stable scatter. Stage chunk to LDS (async LDS-DMA b128 on
// gfx1250, sync b128 fallback), per-thread contiguous segment counting,
// block-wide 8-vector Hillis-Steele scan, LDS-resident cursors, scatter
// scores + (flat>>1) token indices.
// ---------------------------------------------------------------------------
__global__ __launch_bounds__(NTHR) void k_scatter(const int* __restrict__ experts,
                                                  const float* __restrict__ scores,
                                                  const int* __restrict__ offsets,
                                                  const int* __restrict__ bases,
                                                  float* __restrict__ out_scores,
                                                  float* __restrict__ out_tok) {
    __shared__ int sh_exp[CHUNK];                 // 32 KB
    __shared__ int sc[NTHR][NUM_EXPERTS + 1];     // padded rows (9 dw) vs bank conflicts
    __shared__ int blkoff[NUM_EXPERTS];

    const int t = threadIdx.x;
    const int b = blockIdx.x;
    const unsigned base = (unsigned)b * CHUNK;
    const int li0 = t * PER_THREAD;

    // Stage experts chunk into LDS, fully coalesced (16B per lane per op).
    const int4* e4 = (const int4*)(experts + base);
    int4* s4 = (int4*)sh_exp;
#if HAVE_ASYNC_LDS
#pragma unroll
    for (int i = 0; i < CHUNK / 4 / NTHR; ++i) {
        __builtin_amdgcn_global_load_async_to_lds_b128(
            (gv4i_ptr)(e4 + i * NTHR + t),
            (lv4i_ptr)(s4 + i * NTHR + t),
            0, 0);
    }
#else
#pragma unroll
    for (int i = 0; i < CHUNK / 4 / NTHR; ++i) s4[i * NTHR + t] = e4[i * NTHR + t];
#endif

    if (t < NUM_EXPERTS) blkoff[t] = bases[t] + offsets[b * NUM_EXPERTS + t];

    // Warm this thread's score line for the scatter phase (global_prefetch_b8):
    // 32 floats = exactly one 128B line per thread.
    __builtin_prefetch(&scores[base + (unsigned)li0], 0, 0);

#if HAVE_ASYNC_LDS
    __builtin_amdgcn_s_wait_asynccnt(0);   // our async LDS writes landed
#endif
    __syncthreads();                       // ... and are visible block-wide

    // Count experts in this thread's contiguous segment (stable sub-order).
    unsigned lo = 0, hi = 0;
#pragma unroll
    for (int j = 0; j < PER_THREAD; ++j) {
        unsigned e = (unsigned)sh_exp[li0 + j];
        unsigned inc = 1u << ((e & 3u) * 8u);
        if (e & 4u) hi += inc; else lo += inc;
    }
    int cnt[NUM_EXPERTS];
#pragma unroll
    for (int e = 0; e < 4; ++e) {
        cnt[e]     = (int)((lo >> (e * 8)) & 0xffu);
        cnt[e + 4] = (int)((hi >> (e * 8)) & 0xffu);
    }

    // Block-wide inclusive scan across threads on 8-vectors (Hillis-Steele).
    int v[NUM_EXPERTS];
#pragma unroll
    for (int e = 0; e < NUM_EXPERTS; ++e) { v[e] = cnt[e]; sc[t][e] = v[e]; }
    __syncthreads();
    for (int d = 1; d < NTHR; d <<= 1) {
        int u[NUM_EXPERTS];
        if (t >= d) {
#pragma unroll
            for (int e = 0; e < NUM_EXPERTS; ++e) u[e] = sc[t - d][e];
        }
        __syncthreads();
        if (t >= d) {
#pragma unroll
            for (int e = 0; e < NUM_EXPERTS; ++e) v[e] += u[e];
        }
#pragma unroll
        for (int e = 0; e < NUM_EXPERTS; ++e) sc[t][e] = v[e];
        __syncthreads();
    }

    // Final per-thread cursors (global position of this thread's first element
    // of each expert). Own LDS row only: no cross-thread reads after the
    // scan's final barrier, so no extra barrier needed.
#pragma unroll
    for (int e = 0; e < NUM_EXPERTS; ++e) sc[t][e] = blkoff[e] + v[e] - cnt[e];

    // Scatter: contiguous float4 score loads, per-element bucketed stores.
    const float4* sv4 = (const float4*)(scores + base + (unsigned)li0);
#pragma unroll 2
    for (int jj = 0; jj < PER_THREAD / 4; ++jj) {
        float4 s = sv4[jj];
        float sval[4] = {s.x, s.y, s.z, s.w};
#pragma unroll
        for (int k = 0; k < 4; ++k) {
            const int j = jj * 4 + k;
            const int e = sh_exp[li0 + j];
            const int pos = sc[t][e]++;                 // private LDS cursor
            const unsigned flat = base + (unsigned)(li0 + j);
            out_scores[pos] = sval[k];
            out_tok[pos] = (float)(flat >> 1);          // flat / TOP_K, exact (< 2^23)
        }
    }
}

// ---------------------------------------------------------------------------
extern "C" void kernel_launch(void* const* d_in, const int* in_sizes, int n_in,
                              void* d_out, int out_size, void* d_ws, size_t ws_size,
                              hipStream_t stream) {
    const float* scores  = (const float*)d_in[0];   // top_scores, f32, N*TOP_K
    const int*   experts = (const int*)d_in[1];     // selected_experts_indices, i32

    float* out        = (float*)d_out;
    float* out_scores = out;                        // [0, M)
    float* out_tok    = out + (size_t)MTOT;         // [M, 2M)
    float* out_counts = out + (size_t)2 * MTOT;     // [2M, 2M+8)

    int* counts  = (int*)d_ws;                      // NBLK*8 ints
    int* offsets = counts + NBLK * NUM_EXPERTS;     // NBLK*8 ints
    int* bases   = offsets + NBLK * NUM_EXPERTS;    // 8 ints

    k_hist<<<NBLK, NTHR, 0, stream>>>(experts, counts);
    k_scan<<<1, 256, 0, stream>>>(counts, offsets, bases, out_counts);
    k_scatter<<<NBLK, NTHR, 0, stream>>>(experts, scores, offsets, bases,
                                         out_scores, out_tok);
}